// ACloss_50869592655245
// MI455X (gfx1250) — compile-verified
//
#include <hip/hip_runtime.h>
#include <hip/hip_bf16.h>
#include <math.h>

// ---------------- problem constants ----------------
#define HH 512
#define WW 512
#define HW (HH * WW)          // 262144 elements per heatmap
#define NB 8                  // batch
#define NL 19                 // landmarks
#define NHM (NB * NL)         // 152 heatmaps
#define SUB 32                // blocks per heatmap (streaming split)
#define NPART (NHM * SUB)     // 4864 partials
#define CHUNK (HW / SUB)      // 8192 floats per block per tensor
#define THREADS 256
#define F4PT (CHUNK / 4 / THREADS)  // 8 float4 per thread per tensor

typedef __attribute__((ext_vector_type(2))) float v2f;
typedef __attribute__((ext_vector_type(4))) float v4f;   // clang vector -> NT builtin OK
typedef __attribute__((ext_vector_type(8))) float v8f;

// ------------------------------------------------------------------
// Kernel 1: fused streaming pass. Bandwidth-bound: 320MB @ 23.3TB/s.
// Per block: partial L2 sum + per-tensor (max, first-index) over its chunk.
// Non-temporal b128 loads: data is read exactly once and exceeds L2 (192MB).
// ------------------------------------------------------------------
__global__ __launch_bounds__(THREADS) void acloss_stream(
    const float* __restrict__ outp, const float* __restrict__ tgtp,
    float* __restrict__ l2p,
    float* __restrict__ maxO, int* __restrict__ idxO,
    float* __restrict__ maxT, int* __restrict__ idxT)
{
    const int blk = blockIdx.x;
    const int hm  = blk / SUB;
    const int sub = blk % SUB;
    const int tid = threadIdx.x;

    const v4f* o4 = (const v4f*)(outp + (size_t)hm * HW + (size_t)sub * CHUNK);
    const v4f* t4 = (const v4f*)(tgtp + (size_t)hm * HW + (size_t)sub * CHUNK);

    float l2  = 0.0f;
    float bvO = -__builtin_inff(); int biO = 0;
    float bvT = -__builtin_inff(); int biT = 0;
    const int elemBase = sub * CHUNK;

#pragma unroll
    for (int k = 0; k < F4PT; ++k) {
        const int f4 = k * THREADS + tid;            // increasing per thread -> first-occurrence kept
        v4f o = __builtin_nontemporal_load(o4 + f4);
        v4f t = __builtin_nontemporal_load(t4 + f4);
        const int e0 = elemBase + f4 * 4;

        float d0 = o.x - t.x, d1 = o.y - t.y, d2 = o.z - t.z, d3 = o.w - t.w;
        l2 = fmaf(d0, d0, l2); l2 = fmaf(d1, d1, l2);
        l2 = fmaf(d2, d2, l2); l2 = fmaf(d3, d3, l2);

        if (o.x > bvO) { bvO = o.x; biO = e0;     }
        if (o.y > bvO) { bvO = o.y; biO = e0 + 1; }
        if (o.z > bvO) { bvO = o.z; biO = e0 + 2; }
        if (o.w > bvO) { bvO = o.w; biO = e0 + 3; }

        if (t.x > bvT) { bvT = t.x; biT = e0;     }
        if (t.y > bvT) { bvT = t.y; biT = e0 + 1; }
        if (t.z > bvT) { bvT = t.z; biT = e0 + 2; }
        if (t.w > bvT) { bvT = t.w; biT = e0 + 3; }
    }

    __shared__ float sV[THREADS];  __shared__ int sI[THREADS];
    __shared__ float sV2[THREADS]; __shared__ int sI2[THREADS];
    __shared__ float sL[THREADS];
    sV[tid] = bvO; sI[tid] = biO; sV2[tid] = bvT; sI2[tid] = biT; sL[tid] = l2;
    __syncthreads();

    for (int s = THREADS / 2; s > 0; s >>= 1) {
        if (tid < s) {
            sL[tid] += sL[tid + s];
            float v = sV[tid + s]; int i = sI[tid + s];
            if (v > sV[tid] || (v == sV[tid] && i < sI[tid])) { sV[tid] = v; sI[tid] = i; }
            float v2 = sV2[tid + s]; int i2 = sI2[tid + s];
            if (v2 > sV2[tid] || (v2 == sV2[tid] && i2 < sI2[tid])) { sV2[tid] = v2; sI2[tid] = i2; }
        }
        __syncthreads();
    }
    if (tid == 0) {
        l2p[blk]  = sL[0];
        maxO[blk] = sV[0];  idxO[blk] = sI[0];
        maxT[blk] = sV2[0]; idxT[blk] = sI2[0];
    }
}

// ------------------------------------------------------------------
// Kernel 2: single-block finisher.
//   Phase A: deterministic fp64 reduce of L2 partials; finalize 304 argmaxes.
//   Phase B: per-wave (== per-batch, 8 waves of wave32) centered-point grams
//            via V_WMMA_F32_16X16X4_F32 (19 padded to 32, K 2 padded to 4).
//            Angle AND distance both derive from the centered gram:
//            cos_ij = G_ij / sqrt(G_ii G_jj),  d_ij^2 = G_ii + G_jj - 2 G_ij.
// ------------------------------------------------------------------
__global__ __launch_bounds__(THREADS) void acloss_finish(
    const float* __restrict__ l2p,
    const float* __restrict__ maxO, const int* __restrict__ idxO,
    const float* __restrict__ maxT, const int* __restrict__ idxT,
    float* __restrict__ out5)
{
    const int tid = threadIdx.x;

    __shared__ double sD[THREADS];
    __shared__ float  sA[THREADS];
    __shared__ float  sDst[THREADS];
    __shared__ float  sCY[2 * NB * 32];   // [sel][batch][idx<32], zero-padded
    __shared__ float  sCX[2 * NB * 32];
    __shared__ float  sN2[2 * NB * 32];
    __shared__ float  sL2;

    // zero coordinate padding (indices 19..31 must be 0 for the gram tiles)
    for (int i = tid; i < 2 * NB * 32; i += THREADS) { sCY[i] = 0.0f; sCX[i] = 0.0f; }

    // ---- L2 partial reduce (fp64, fixed tree order -> deterministic) ----
    double acc = 0.0;
    for (int i = tid; i < NPART; i += THREADS) acc += (double)l2p[i];
    sD[tid] = acc;
    __syncthreads();
    for (int s = THREADS / 2; s > 0; s >>= 1) {
        if (tid < s) sD[tid] += sD[tid + s];
        __syncthreads();
    }
    if (tid == 0) sL2 = (float)(sD[0] / ((double)NB * (double)NL * (double)HW));

    // ---- finalize argmax -> centered peak coordinates ----
    for (int h = tid; h < 2 * NHM; h += THREADS) {
        const int sel = h / NHM;       // 0 = output, 1 = target
        const int hm  = h % NHM;
        const float* mv = sel ? maxT : maxO;
        const int*   mi = sel ? idxT : idxO;
        float bv = -__builtin_inff(); int bi = 0;
        for (int s = 0; s < SUB; ++s) {
            float v = mv[hm * SUB + s]; int i2 = mi[hm * SUB + s];
            if (v > bv || (v == bv && i2 < bi)) { bv = v; bi = i2; }
        }
        const int y = bi / WW, x = bi % WW;
        const int b = hm / NL, l = hm % NL;
        const int o = sel * NB * 32 + b * 32 + l;
        sCY[o] = (float)y - (float)(HH / 2);   // centered about (256, 256)
        sCX[o] = (float)x - (float)(WW / 2);
    }
    __syncthreads();
    for (int i = tid; i < 2 * NB * 32; i += THREADS)
        sN2[i] = sCY[i] * sCY[i] + sCX[i] * sCX[i];
    __syncthreads();

    // ---- WMMA gram phase: wave w <-> batch w, full EXEC at every wmma ----
    const int wave = tid >> 5;
    const int lane = tid & 31;
    const int lo   = lane & 15;
    const bool hiHalf = (lane >= 16);
    const int baseO = 0 * NB * 32 + wave * 32;
    const int baseT = 1 * NB * 32 + wave * 32;

    float sumAng = 0.0f, sumDst = 0.0f;

    for (int mo = 0; mo < 32; mo += 16) {
        for (int no = 0; no < 32; no += 16) {
            // A (16x4 f32): lanes 0-15 carry K=0,1 (y,x); lanes 16-31 carry K=2,3 (zero pad)
            // B (4x16 f32) = A^T of the column tile -> same per-lane packing.
            v2f aO, bO, aT, bT;
            if (!hiHalf) {
                aO = (v2f){ sCY[baseO + mo + lo], sCX[baseO + mo + lo] };
                bO = (v2f){ sCY[baseO + no + lo], sCX[baseO + no + lo] };
                aT = (v2f){ sCY[baseT + mo + lo], sCX[baseT + mo + lo] };
                bT = (v2f){ sCY[baseT + no + lo], sCX[baseT + no + lo] };
            } else {
                aO = (v2f){0.0f, 0.0f}; bO = aO; aT = aO; bT = aO;
            }
            v8f c = {};
            v8f dO = __builtin_amdgcn_wmma_f32_16x16x4_f32(
                false, aO, false, bO, (short)0, c, false, false);
            v8f dT = __builtin_amdgcn_wmma_f32_16x16x4_f32(
                false, aT, false, bT, (short)0, c, false, false);

#pragma unroll
            for (int r = 0; r < 8; ++r) {
                const int i = mo + r + (hiHalf ? 8 : 0);  // C/D layout: VGPR r, lane half
                const int j = no + lo;
                if (i < NL && j < NL) {
                    const float gO = dO[r], gT = dT[r];
                    const float n2Oi = sN2[baseO + i], n2Oj = sN2[baseO + j];
                    const float n2Ti = sN2[baseT + i], n2Tj = sN2[baseT + j];

                    float angO = 0.0f, angT = 0.0f;           // nan_to_num(0) path
                    const float pO = n2Oi * n2Oj;
                    if (pO > 0.0f) {
                        float cs = gO / sqrtf(pO);
                        cs = fminf(1.0f, fmaxf(-1.0f, cs));
                        angO = acosf(cs);
                    }
                    const float pT = n2Ti * n2Tj;
                    if (pT > 0.0f) {
                        float cs = gT / sqrtf(pT);
                        cs = fminf(1.0f, fmaxf(-1.0f, cs));
                        angT = acosf(cs);
                    }
                    sumAng += fabsf(angO - angT);

                    const float dO2 = fmaxf(n2Oi + n2Oj - 2.0f * gO, 0.0f);
                    const float dT2 = fmaxf(n2Ti + n2Tj - 2.0f * gT, 0.0f);
                    sumDst += fabsf(sqrtf(dO2) - sqrtf(dT2));
                }
            }
        }
    }

    sA[tid] = sumAng; sDst[tid] = sumDst;
    __syncthreads();
    for (int s = THREADS / 2; s > 0; s >>= 1) {
        if (tid < s) { sA[tid] += sA[tid + s]; sDst[tid] += sDst[tid + s]; }
        __syncthreads();
    }

    if (tid == 0) {
        // sum_b mean_b(|.|) == total_sum / (L*L)
        const float angle_loss = sA[0]   / (float)(NL * NL);
        const float dist_loss  = sDst[0] / (float)(NL * NL);
        const float l2l = sL2;
        const float w   = 1.0f + angle_loss + logf(dist_loss + 1e-10f);
        out5[0] = l2l * w;
        out5[1] = l2l;
        out5[2] = w;
        out5[3] = angle_loss;
        out5[4] = dist_loss;
    }
}

extern "C" void kernel_launch(void* const* d_in, const int* in_sizes, int n_in,
                              void* d_out, int out_size, void* d_ws, size_t ws_size,
                              hipStream_t stream) {
    const float* outp = (const float*)d_in[0];
    const float* tgtp = (const float*)d_in[1];
    float* out5 = (float*)d_out;

    float* l2p  = (float*)d_ws;
    float* maxO = l2p + NPART;
    int*   idxO = (int*)(maxO + NPART);
    float* maxT = (float*)(idxO + NPART);
    int*   idxT = (int*)(maxT + NPART);

    acloss_stream<<<NPART, THREADS, 0, stream>>>(outp, tgtp, l2p, maxO, idxO, maxT, idxT);
    acloss_finish<<<1, THREADS, 0, stream>>>(l2p, maxO, idxO, maxT, idxT, out5);
}